// Dct2d_42425686949911
// MI455X (gfx1250) — compile-verified
//
#include <hip/hip_runtime.h>
#include <hip/hip_bf16.h>

// CDNA5 / gfx1250: wave32, WMMA f32 16x16x4.
typedef __attribute__((ext_vector_type(2))) float v2f;
typedef __attribute__((ext_vector_type(8))) float v8f;

#define DCT_W 1024            // reference fixes H = W = 1024
#define TILE 16               // one 16x16 tile = four 8x8 DCT blocks
#define TILES_X (DCT_W / TILE)            // 64
#define TILES_PER_IMG (TILES_X * TILES_X) // 4096
#define WAVES_PER_BLOCK 8
#define LDS_STRIDE 18         // even -> 8B-aligned float2 rows, bank-conflict free

__global__ __launch_bounds__(WAVES_PER_BLOCK * 32)
void dct8x8_wmma_kernel(const float* __restrict__ x,
                        const float* __restrict__ A,
                        float* __restrict__ out,
                        int total_tiles)
{
    __shared__ float lds[WAVES_PER_BLOCK][TILE * LDS_STRIDE];

    const int lane = threadIdx.x & 31;
    const int wave = threadIdx.x >> 5;
    const int tile = blockIdx.x * WAVES_PER_BLOCK + wave;
    if (tile >= total_tiles) return;   // uniform per wave -> EXEC stays all-1s for WMMA

    const int img = tile >> 12;        // / TILES_PER_IMG (4096)
    const int rem = tile & (TILES_PER_IMG - 1);
    const int ty  = rem >> 6;          // / TILES_X
    const int tx  = rem & (TILES_X - 1);

    const size_t W    = (size_t)DCT_W;
    const size_t base = (size_t)img * (size_t)TILES_PER_IMG * (TILE * TILE)
                      + (size_t)ty * TILE * W
                      + (size_t)tx * TILE;

    const int m  = lane & 15;          // row (A-op) / col (B-op, C/D)
    const int hi = lane >> 4;          // lane-half selects K+2 (A/B) or M+8 (C/D)

    // --- Â = diag(A, A) fragments -------------------------------------------
    // A-operand slice k of Â: lane(m,hi) comp r = Â[m][4k + r + 2*hi].
    // The SAME fragment, fed as a B-operand, represents Âᵀ (mapping coincides).
    v2f af[4];
#pragma unroll
    for (int k = 0; k < 4; ++k) {
        const int k0 = 4 * k + 2 * hi;
        const bool nz = ((m ^ k0) & 8) == 0;      // block-diagonal mask
        v2f t;
        t.x = nz ? A[(m & 7) * 8 + (k0 & 7)]       : 0.0f;
        t.y = nz ? A[(m & 7) * 8 + ((k0 + 1) & 7)] : 0.0f;
        af[k] = t;
    }

    // --- load X directly in B-operand layout --------------------------------
    // B slice k, comp r = X[4k + r + 2*hi][m]  (tile-local). Each half-wave
    // reads a contiguous 64B row segment -> coalesced global_load_b32.
    const float* xt = x + base;
    v2f xb[4];
#pragma unroll
    for (int k = 0; k < 4; ++k) {
        const int r0 = 4 * k + 2 * hi;
        v2f t;
        t.x = xt[(size_t)r0 * W + m];
        t.y = xt[(size_t)(r0 + 1) * W + m];
        xb[k] = t;
    }

    // --- GEMM 1: U = Â · X  (four chained 16x16x4 f32 WMMAs) ----------------
    v8f u = {};
#pragma unroll
    for (int k = 0; k < 4; ++k)
        u = __builtin_amdgcn_wmma_f32_16x16x4_f32(
                false, af[k], false, xb[k], (short)0, u, false, false);

    // --- C/D layout -> A-operand layout via LDS bounce ----------------------
    // C/D: VGPR r = U[r + 8*hi][m]. Write rows, read back float2 per lane.
    float* myl = lds[wave];
#pragma unroll
    for (int r = 0; r < 8; ++r)
        myl[(r + 8 * hi) * LDS_STRIDE + m] = u[r];
    // same-wave DS RAW: compiler inserts s_wait_dscnt

    v2f ua[4];
#pragma unroll
    for (int k = 0; k < 4; ++k) {
        const v2f* p = (const v2f*)(myl + m * LDS_STRIDE + 4 * k + 2 * hi);
        ua[k] = *p;                    // ds_load_b64, 8B aligned (stride even)
    }

    // --- GEMM 2: Y = U · Âᵀ  (af[] reused as the B-operand of Âᵀ) -----------
    v8f y = {};
#pragma unroll
    for (int k = 0; k < 4; ++k)
        y = __builtin_amdgcn_wmma_f32_16x16x4_f32(
                false, ua[k], false, af[k], (short)0, y, false, false);

    // --- store Y straight from C/D layout (coalesced) -----------------------
    float* ot = out + base;
#pragma unroll
    for (int r = 0; r < 8; ++r)
        ot[(size_t)(r + 8 * hi) * W + m] = y[r];
}

extern "C" void kernel_launch(void* const* d_in, const int* in_sizes, int n_in,
                              void* d_out, int out_size, void* d_ws, size_t ws_size,
                              hipStream_t stream) {
    const float* x = (const float*)d_in[0];   // (64,1,1024,1024) fp32
    const float* A = (const float*)d_in[1];   // (8,8) fp32 DCT basis
    float* out     = (float*)d_out;

    const int n_elems     = in_sizes[0];            // 67,108,864
    const int total_tiles = n_elems / (TILE * TILE);// 262,144 waves
    const int blocks = (total_tiles + WAVES_PER_BLOCK - 1) / WAVES_PER_BLOCK;

    dct8x8_wmma_kernel<<<blocks, WAVES_PER_BLOCK * 32, 0, stream>>>(
        x, A, out, total_tiles);
}